// mask_frames_69767448756538
// MI455X (gfx1250) — compile-verified
//
#include <hip/hip_runtime.h>
#include <hip/hip_bf16.h>
#include <stdint.h>

// Problem constants (from reference): frames (B,T,H,W,C) fp32
#define B_  4
#define T_  32
#define H_  224
#define W_  224
#define C_  3
// n1 = int(B*T*H*W * 0.2 * 0.8 // 13005) = 79 ; n2 = int(... * 0.1 // 13005) = 9
#define N1  79
#define N2  9
#define MAXE1 80   // capacity >= N1
#define MAXE2 16   // capacity >= N2

// Workspace layout (uint32 indices). Total ~50 KB.
#define WS_CNT1 0                         // 128 counts (R1 boxes per (b,t))
#define WS_CNT2 128                       // 128 counts (R2 boxes per (b,t))
#define WS_E1   256                       // 128 * MAXE1 packed R1 entries
#define WS_E2   (WS_E1 + 128 * MAXE1)     // 128 * MAXE2 packed R2 entries
#define WS_TOK  (WS_E2 + 128 * MAXE2)     // 3 floats: random_token

#if defined(__has_builtin)
#if __has_builtin(__builtin_amdgcn_global_load_async_to_lds_b32) && \
    __has_builtin(__builtin_amdgcn_s_wait_asynccnt)
#define HAVE_ASYNC_LDS 1
#endif
#endif

typedef __attribute__((address_space(1))) int gas_int;  // global AS ('__device__ int *')
typedef __attribute__((address_space(3))) int las_int;  // LDS AS    ('__shared__ int *')

// ---------------------------------------------------------------------------
// Kernel A: one thread per (b,t). Build per-(b,t) packed box lists, M, token.
// ---------------------------------------------------------------------------
__global__ void __launch_bounds__(128)
build_boxes_kernel(const float* __restrict__ frames,
                   const int* __restrict__ boxb, const int* __restrict__ boxt,
                   const int* __restrict__ boxh, const int* __restrict__ boxw,
                   const int* __restrict__ rb, const int* __restrict__ rt,
                   const int* __restrict__ rh, const int* __restrict__ rw,
                   uint32_t* __restrict__ ws, float* __restrict__ Mout) {
    const int p = threadIdx.x;            // p = b*T + t, 0..127
    if (p >= B_ * T_) return;
    const int myb = p / T_;
    const int myt = p % T_;

    uint32_t c1 = 0, c2 = 0;
    bool any = false;
    for (int i = 0; i < N1 + 2 * N2; ++i) {
        if (boxb[i] != myb) continue;
        const int ti  = boxt[i];
        const int lot = max(ti - 2, 0);
        const int hit = min(ti + 2, T_ - 1);          // torch-style: bound uses size-1
        if (myt < lot || myt >= hit) continue;
        any = true;                                    // h/w intervals are never empty
        if (i >= N1 + N2) continue;                    // R3 contributes to M only
        const int hi_ = boxh[i], wi_ = boxw[i];
        const uint32_t loh = (uint32_t)max(hi_ - 25, 0);
        const uint32_t hih = (uint32_t)min(hi_ + 25, H_ - 1);
        const uint32_t low = (uint32_t)max(wi_ - 25, 0);
        const uint32_t hiw = (uint32_t)min(wi_ + 25, W_ - 1);
        const uint32_t e = (loh << 24) | (hih << 16) | (low << 8) | hiw;
        if (i < N1) ws[WS_E1 + p * MAXE1 + (c1++)] = e;
        else        ws[WS_E2 + p * MAXE2 + (c2++)] = e;
    }
    ws[WS_CNT1 + p] = c1;
    ws[WS_CNT2 + p] = c2;
    Mout[p] = any ? 1.0f : 0.0f;

    if (p == 0) {
        const size_t ridx =
            ((((size_t)rb[0] * T_ + rt[0]) * H_ + rh[0]) * W_ + rw[0]) * C_;
        float* tok = (float*)&ws[WS_TOK];
        tok[0] = frames[ridx + 0];
        tok[1] = frames[ridx + 1];
        tok[2] = frames[ridx + 2];
    }
}

// ---------------------------------------------------------------------------
// Kernel B: streaming blend. Block = 4 H-rows of one (b,t); thread = 4 pixels
// (= 3 x float4). Box lists staged in LDS via async global->LDS copy.
// ---------------------------------------------------------------------------
__global__ void __launch_bounds__(224)
mask_frames_kernel(const float* __restrict__ frames,
                   const float* __restrict__ mask_token,
                   const uint32_t* __restrict__ ws,
                   float* __restrict__ out) {
    __shared__ uint32_t s_e1[MAXE1];
    __shared__ uint32_t s_e2[MAXE2];

    const int tid = threadIdx.x;                     // 0..223
    const int bt  = blockIdx.x / (H_ / 4);           // (b*T + t), uniform per block
    const int h0  = (blockIdx.x % (H_ / 4)) * 4;
    const int hh  = h0 + tid / (W_ / 4);
    const int w0  = (tid % (W_ / 4)) * 4;
    const size_t base = (((size_t)bt * H_ + hh) * W_ + w0) * C_;

    // Kick the HBM stream early (global_prefetch_b8) while we set up LDS.
    __builtin_prefetch(frames + base, 0, 1);

    const uint32_t cnt1 = ws[WS_CNT1 + bt];
    const uint32_t cnt2 = ws[WS_CNT2 + bt];
    const uint32_t* g1 = ws + WS_E1 + (size_t)bt * MAXE1;
    const uint32_t* g2 = ws + WS_E2 + (size_t)bt * MAXE2;

#if HAVE_ASYNC_LDS
    // CDNA5 async global->LDS copy; tracked by ASYNCcnt, no VGPR staging.
    if (tid < MAXE1)
        __builtin_amdgcn_global_load_async_to_lds_b32(
            (gas_int*)(g1 + tid), (las_int*)&s_e1[tid], 0, 0);
    if (tid < MAXE2)
        __builtin_amdgcn_global_load_async_to_lds_b32(
            (gas_int*)(g2 + tid), (las_int*)&s_e2[tid], 0, 0);
    __builtin_amdgcn_s_wait_asynccnt(0);
#else
    if (tid < MAXE1) s_e1[tid] = g1[tid];
    if (tid < MAXE2) s_e2[tid] = g2[tid];
#endif
    __syncthreads();

    // 4 pixels = 12 floats = 3 x b128 loads.
    const float4* fp = (const float4*)(frames + base);
    float4 A = fp[0], Bv = fp[1], Cv = fp[2];

    // Per-pixel coverage masks (bit px = pixel w0+px covered).
    uint32_t m1 = 0, m2 = 0;
    for (uint32_t j = 0; j < cnt1; ++j) {
        const uint32_t e = s_e1[j];
        const int loh = (int)(e >> 24), hih = (int)((e >> 16) & 255u);
        if (hh < loh || hh >= hih) continue;
        const int low = (int)((e >> 8) & 255u), hiw = (int)(e & 255u);
        const int s  = max(low - w0, 0);
        const int en = min(hiw - w0, 4);
        if (s < en) m1 |= ((1u << en) - 1u) & ~((1u << s) - 1u);
    }
    for (uint32_t j = 0; j < cnt2; ++j) {
        const uint32_t e = s_e2[j];
        const int loh = (int)(e >> 24), hih = (int)((e >> 16) & 255u);
        if (hh < loh || hh >= hih) continue;
        const int low = (int)((e >> 8) & 255u), hiw = (int)(e & 255u);
        const int s  = max(low - w0, 0);
        const int en = min(hiw - w0, 4);
        if (s < en) m2 |= ((1u << en) - 1u) & ~((1u << s) - 1u);
    }

    float v[12] = {A.x, A.y, A.z, A.w, Bv.x, Bv.y, Bv.z, Bv.w,
                   Cv.x, Cv.y, Cv.z, Cv.w};

    if (m1 | m2) {                         // rare path: blend tokens in
        const float* rtokp = (const float*)(ws + WS_TOK);
        const float mt[3] = {mask_token[0], mask_token[1], mask_token[2]};
        const float rt[3] = {rtokp[0], rtokp[1], rtokp[2]};
#pragma unroll
        for (int k = 0; k < 12; ++k) {
            const int px = k / 3, c = k % 3;
            if ((m2 >> px) & 1u)      v[k] = rt[c];   // R2 wins over R1
            else if ((m1 >> px) & 1u) v[k] = mt[c];
        }
    }

    float4* op = (float4*)(out + base);
    op[0] = make_float4(v[0], v[1], v[2],  v[3]);
    op[1] = make_float4(v[4], v[5], v[6],  v[7]);
    op[2] = make_float4(v[8], v[9], v[10], v[11]);
}

// ---------------------------------------------------------------------------
extern "C" void kernel_launch(void* const* d_in, const int* in_sizes, int n_in,
                              void* d_out, int out_size, void* d_ws, size_t ws_size,
                              hipStream_t stream) {
    const float* frames     = (const float*)d_in[0];
    const float* mask_token = (const float*)d_in[1];
    const int*   boxb = (const int*)d_in[2];
    const int*   boxt = (const int*)d_in[3];
    const int*   boxh = (const int*)d_in[4];
    const int*   boxw = (const int*)d_in[5];
    const int*   rb   = (const int*)d_in[6];
    const int*   rt   = (const int*)d_in[7];
    const int*   rh   = (const int*)d_in[8];
    const int*   rw   = (const int*)d_in[9];

    float*    out  = (float*)d_out;
    float*    Mout = out + (size_t)B_ * T_ * H_ * W_ * C_;   // M appended after out
    uint32_t* ws   = (uint32_t*)d_ws;

    build_boxes_kernel<<<1, 128, 0, stream>>>(frames, boxb, boxt, boxh, boxw,
                                              rb, rt, rh, rw, ws, Mout);

    const int blocks = B_ * T_ * (H_ / 4);   // 7168
    mask_frames_kernel<<<blocks, 224, 0, stream>>>(frames, mask_token, ws, out);
}